// ConvAttention_36799279792725
// MI455X (gfx1250) — compile-verified
//
#include <hip/hip_runtime.h>
#include <math.h>

typedef float v2f __attribute__((ext_vector_type(2)));
typedef float v8f __attribute__((ext_vector_type(8)));

#define TBM 64
#define TBN 128
#define TKC 16
// 128 threads = 4 waves in a 2x2 grid; each wave computes a 32x64 tile
// as 2(M) x 4(N) V_WMMA_F32_16X16X4_F32 accumulators (A frags reused 4x).

// AMODE: 0 = A[m*lda+k] (weights), 1 = A[k*lda+m] (q^T for scores)
// BMODE: 0 = B[k*ldb+n], 1 = implicit im2col conv(k=3,pad=1): k=(ci,dk), B[ci*ldb + n+dk-1]
// EPI:   0 = +bias, 1 = relu(+bias), 2 = score: -5e-4*(|a_row|^2+|b_col|^2-2*acc)

template<int AMODE>
__device__ __forceinline__ const float* addrA(const float* A, int gm, int gk, int lda) {
    return (AMODE == 0) ? &A[(long)gm * lda + gk] : &A[(long)gk * lda + gm];
}

template<int AMODE, int BMODE, int EPI>
__global__ __launch_bounds__(128)
void gemm_wmma_f32(const float* __restrict__ A, long sA,
                   const float* __restrict__ Bm, long sB,
                   const float* __restrict__ bias,
                   float* __restrict__ C, long sC,
                   int M, int N, int K, int lda, int ldb, int ldc)
{
    __shared__ float As[TKC * TBM];   // [k][m]
    __shared__ float Bs[TKC * TBN];   // [k][n]

    const int tid  = threadIdx.x;
    const int lane = tid & 31;
    const int wave = tid >> 5;
    const int h    = lane >> 4;    // K-pair half within a step of 4 (K: 2h, 2h+1)
    const int ln   = lane & 15;    // row (A) / col (B/C) within 16-wide tile
    const int wm   = (wave >> 1) * 32;   // wave M offset: 0 / 32
    const int wn   = (wave & 1) * 64;    // wave N offset: 0 / 64
    const int bM   = blockIdx.y * TBM;
    const int bN   = blockIdx.x * TBN;

    A  += (long)blockIdx.z * sA;
    Bm += (long)blockIdx.z * sB;
    C  += (long)blockIdx.z * sC;

    v8f acc[2][4];
    for (int i = 0; i < 2; ++i)
        for (int j = 0; j < 4; ++j)
            for (int r = 0; r < 8; ++r) acc[i][j][r] = 0.0f;

    float sqA[2] = {0.0f, 0.0f};                 // partial |row|^2 (EPI==2)
    float sqB[4] = {0.0f, 0.0f, 0.0f, 0.0f};     // partial |col|^2 (EPI==2)

    for (int k0 = 0; k0 < K; k0 += TKC) {
        // cooperative global -> LDS (zero-padded), prefetch next chunk to cache
        for (int i = tid; i < TKC * TBM; i += 128) {
            int kk = i >> 6, m = i & 63;
            int gm = bM + m, gk = k0 + kk;
            float v = 0.0f;
            if (gm < M && gk < K) {
                v = *addrA<AMODE>(A, gm, gk, lda);
                if (gk + TKC < K)
                    __builtin_prefetch(addrA<AMODE>(A, gm, gk + TKC, lda), 0, 3);
            }
            As[i] = v;
        }
        for (int i = tid; i < TKC * TBN; i += 128) {
            int kk = i >> 7, n = i & 127;
            int gn = bN + n, gk = k0 + kk;
            float v = 0.0f;
            if (gn < N && gk < K) {
                if (BMODE == 0) {
                    v = Bm[(long)gk * ldb + gn];
                    if (gk + TKC < K)
                        __builtin_prefetch(&Bm[(long)(gk + TKC) * ldb + gn], 0, 3);
                } else {
                    int ci = gk / 3, dk = gk - 3 * ci;
                    int t = gn + dk - 1;               // pad = 1
                    if (t >= 0 && t < ldb) v = Bm[(long)ci * ldb + t];
                    int gk2 = gk + TKC;
                    if (gk2 < K) {
                        int ci2 = gk2 / 3, dk2 = gk2 - 3 * ci2;
                        int t2 = gn + dk2 - 1;
                        if (t2 >= 0 && t2 < ldb)
                            __builtin_prefetch(&Bm[(long)ci2 * ldb + t2], 0, 3);
                    }
                }
            }
            Bs[i] = v;
        }
        __syncthreads();

        for (int ks = 0; ks < TKC; ks += 4) {
            int kk = ks + 2 * h;
            v2f af[2], bf[4];
            for (int mt = 0; mt < 2; ++mt) {
                int mm = wm + mt * 16 + ln;
                af[mt].x = As[kk * TBM + mm];
                af[mt].y = As[(kk + 1) * TBM + mm];
                if (EPI == 2) sqA[mt] += af[mt].x * af[mt].x + af[mt].y * af[mt].y;
            }
            for (int nt = 0; nt < 4; ++nt) {
                int nn = wn + nt * 16 + ln;
                bf[nt].x = Bs[kk * TBN + nn];
                bf[nt].y = Bs[(kk + 1) * TBN + nn];
                if (EPI == 2) sqB[nt] += bf[nt].x * bf[nt].x + bf[nt].y * bf[nt].y;
            }
            for (int mt = 0; mt < 2; ++mt)
                for (int nt = 0; nt < 4; ++nt)
                    acc[mt][nt] = __builtin_amdgcn_wmma_f32_16x16x4_f32(
                        false, af[mt], false, bf[nt], (short)0, acc[mt][nt],
                        false, false);
        }
        __syncthreads();
    }

    float q2f[2], k2f[4];
    if (EPI == 2) {
        for (int mt = 0; mt < 2; ++mt) { float s = sqA[mt]; q2f[mt] = s + __shfl_xor(s, 16, 32); }
        for (int nt = 0; nt < 4; ++nt) { float s = sqB[nt]; k2f[nt] = s + __shfl_xor(s, 16, 32); }
    }

    for (int mt = 0; mt < 2; ++mt)
        for (int nt = 0; nt < 4; ++nt)
            for (int r = 0; r < 8; ++r) {
                int gm = bM + wm + mt * 16 + h * 8 + r;   // C row layout: VGPR r = row r + 8h
                int gn = bN + wn + nt * 16 + ln;
                float v = acc[mt][nt][r];
                if (EPI == 2) {
                    float q2 = __shfl(q2f[mt], r + 8 * h, 32);  // lane (r+8h) owns row r+8h
                    v = -0.0005f * (q2 + k2f[nt] - 2.0f * v);
                    if (gm < M && gn < N) C[(long)gm * ldc + gn] = v;
                } else {
                    if (gm < M && gn < N) {
                        v += bias[gm];
                        if (EPI == 1) v = fmaxf(v, 0.0f);
                        C[(long)gm * ldc + gn] = v;
                    }
                }
            }
}

// One block per (b, t1) row of 512 scores. In-place: raw scores -> attn_logprob,
// and masked softmax -> attn.
__global__ __launch_bounds__(256)
void softmax_rows(float* __restrict__ lp, float* __restrict__ attn,
                  const float* __restrict__ prior,
                  const unsigned char* __restrict__ mask)
{
    const int T2 = 512;
    const int row = blockIdx.x;          // b*2048 + t1
    const int b   = row >> 11;
    const int tid = threadIdx.x;
    float* srow = lp + (size_t)row * T2;
    const float* prow = prior + (size_t)row * T2;
    const unsigned char* mrow = mask + (size_t)b * T2;

    __shared__ float red[256];
    float s0 = srow[tid], s1 = srow[tid + 256];

    float m = fmaxf(s0, s1);
    red[tid] = m; __syncthreads();
    for (int off = 128; off > 0; off >>= 1) {
        if (tid < off) red[tid] = fmaxf(red[tid], red[tid + off]);
        __syncthreads();
    }
    float rowmax = red[0]; __syncthreads();

    float e = expf(s0 - rowmax) + expf(s1 - rowmax);
    red[tid] = e; __syncthreads();
    for (int off = 128; off > 0; off >>= 1) {
        if (tid < off) red[tid] += red[tid + off];
        __syncthreads();
    }
    float lse = logf(red[0]) + rowmax; __syncthreads();

    float lp0 = s0 - lse + logf(prow[tid] + 1e-8f);
    float lp1 = s1 - lse + logf(prow[tid + 256] + 1e-8f);
    srow[tid] = lp0;
    srow[tid + 256] = lp1;

    float g0 = mrow[tid]       ? -INFINITY : lp0;
    float g1 = mrow[tid + 256] ? -INFINITY : lp1;
    float m2 = fmaxf(g0, g1);
    red[tid] = m2; __syncthreads();
    for (int off = 128; off > 0; off >>= 1) {
        if (tid < off) red[tid] = fmaxf(red[tid], red[tid + off]);
        __syncthreads();
    }
    float rmax2 = red[0]; __syncthreads();

    float p0 = expf(g0 - rmax2), p1 = expf(g1 - rmax2);
    red[tid] = p0 + p1; __syncthreads();
    for (int off = 128; off > 0; off >>= 1) {
        if (tid < off) red[tid] += red[tid + off];
        __syncthreads();
    }
    float inv = 1.0f / red[0];
    attn[(size_t)row * T2 + tid]       = p0 * inv;
    attn[(size_t)row * T2 + tid + 256] = p1 * inv;
}

extern "C" void kernel_launch(void* const* d_in, const int* in_sizes, int n_in,
                              void* d_out, int out_size, void* d_ws, size_t ws_size,
                              hipStream_t stream)
{
    const float* queries = (const float*)d_in[0];   // [32,100,2048]
    const float* keys    = (const float*)d_in[1];   // [32,512,512]
    // d_in[2] = query_lens: unused by the reference outputs
    const unsigned char* mask = (const unsigned char*)d_in[3];  // [32,512,1] bool
    const float* prior = (const float*)d_in[4];     // [32,2048,512]
    const float* kW1 = (const float*)d_in[5];
    const float* kb1 = (const float*)d_in[6];
    const float* kW2 = (const float*)d_in[7];
    const float* kb2 = (const float*)d_in[8];
    const float* qW1 = (const float*)d_in[9];
    const float* qb1 = (const float*)d_in[10];
    const float* qW2 = (const float*)d_in[11];
    const float* qb2 = (const float*)d_in[12];
    const float* qW3 = (const float*)d_in[13];
    const float* qb3 = (const float*)d_in[14];

    const int Bz = 32, T1 = 2048, T2 = 512, Catt = 100;
    float* O = (float*)d_out;
    const size_t NT = (size_t)Bz * T1 * T2;         // 33,554,432 per output
    float* attn = O;                                 // output 0
    float* lp   = O + NT;                            // output 1 (scores staged here)

    // Scratch arena inside the attn half (dead until softmax_rows writes it):
    float* q     = O;                                // 32*100*2048 = 6,553,600 f
    float* kmat  = O + 6553600;                      // 32*100*512  = 1,638,400 f
    float* arena = O + 8192000;
    float* qHid  = arena;                            // 32*200*2048 = 13,107,200 f
    float* qMid  = arena + 13107200;                 // ends 27,852,800 < 33,554,432
    float* kHid  = arena;                            // 32*1024*512 = 16,777,216 f (after q chain)

    dim3 blk(128);

    // query proj: relu(conv3) -> relu(1x1) -> 1x1
    gemm_wmma_f32<0,1,1><<<dim3(T1/TBN, (200+TBM-1)/TBM, Bz), blk, 0, stream>>>(
        qW1, 0, queries, (long)100*T1, qb1, qHid, (long)200*T1,
        200, T1, 300, 300, T1, T1);
    gemm_wmma_f32<0,0,1><<<dim3(T1/TBN, (100+TBM-1)/TBM, Bz), blk, 0, stream>>>(
        qW2, 0, qHid, (long)200*T1, qb2, qMid, (long)100*T1,
        100, T1, 200, 200, T1, T1);
    gemm_wmma_f32<0,0,0><<<dim3(T1/TBN, (100+TBM-1)/TBM, Bz), blk, 0, stream>>>(
        qW3, 0, qMid, (long)100*T1, qb3, q, (long)100*T1,
        100, T1, 100, 100, T1, T1);

    // key proj: relu(conv3) -> 1x1
    gemm_wmma_f32<0,1,1><<<dim3(T2/TBN, 1024/TBM, Bz), blk, 0, stream>>>(
        kW1, 0, keys, (long)512*T2, kb1, kHid, (long)1024*T2,
        1024, T2, 1536, 1536, T2, T2);
    gemm_wmma_f32<0,0,0><<<dim3(T2/TBN, (100+TBM-1)/TBM, Bz), blk, 0, stream>>>(
        kW2, 0, kHid, (long)1024*T2, kb2, kmat, (long)100*T2,
        100, T2, 1024, 1024, T2, T2);

    // scores[b,t1,t2] = -5e-4 * (|q|^2 + |k|^2 - 2 q^T k), fused epilogue
    gemm_wmma_f32<1,0,2><<<dim3(T2/TBN, T1/TBM, Bz), blk, 0, stream>>>(
        q, (long)Catt*T1, kmat, (long)Catt*T2, nullptr, lp, (long)T1*T2,
        T1, T2, Catt, T1, T2, T2);

    // row softmaxes: writes attn_logprob in place and attn
    softmax_rows<<<dim3(Bz * T1), dim3(256), 0, stream>>>(lp, attn, prior, mask);
}